// RATMAttention_39067022524480
// MI455X (gfx1250) — compile-verified
//
#include <hip/hip_runtime.h>

// Problem constants (match reference)
constexpr int kH  = 224;
constexpr int kW  = 224;
constexpr int kGH = 32;
constexpr int kGW = 32;
constexpr int kB  = 16;
constexpr int kC  = 64;
constexpr int kG  = 4;

typedef float v2f __attribute__((ext_vector_type(2)));
typedef float v8f __attribute__((ext_vector_type(8)));
typedef int   v4i_ __attribute__((vector_size(16)));   // matches builtin param type

// Stride 228: rows 16B-aligned (228*4=912 bytes) for async b128 LDS writes.
// Phase-2 T reads: lane banks = {4k} vs {4k+2} across lane halves -> conflict-free.
#define XS_STRIDE 228

static __device__ __forceinline__ v8f wmma_f32_k4(v2f a, v2f b, v8f c) {
    // D = A(16x4,f32) * B(4x16,f32) + C(16x16,f32)
    return __builtin_amdgcn_wmma_f32_16x16x4_f32(
        /*neg_a=*/false, a, /*neg_b=*/false, b,
        /*c_mod=*/(short)0, c, /*reuse_a=*/false, /*reuse_b=*/false);
}

// 16-byte global -> LDS copy. Prefer the CDNA5 async DMA path (ASYNCcnt-tracked,
// no VGPR round trip); fall back to ld.b128 + ds_store if the builtin is absent.
static __device__ __forceinline__ void cp16_g2l(const float* __restrict__ g, float* l) {
#if __has_builtin(__builtin_amdgcn_global_load_async_to_lds_b128)
    __builtin_amdgcn_global_load_async_to_lds_b128(
        (__attribute__((address_space(1))) v4i_*)(float*)g,
        (__attribute__((address_space(3))) v4i_*)l,
        /*offset=*/0, /*cpol=*/0);
#else
    float4 q = *(const float4*)g;
    l[0] = q.x; l[1] = q.y; l[2] = q.z; l[3] = q.w;
#endif
}

static __device__ __forceinline__ void async_wait0() {
#if __has_builtin(__builtin_amdgcn_global_load_async_to_lds_b128)
#if __has_builtin(__builtin_amdgcn_s_wait_asynccnt)
    __builtin_amdgcn_s_wait_asynccnt(0);
#else
    asm volatile("s_wait_asynccnt 0x0" ::: "memory");
#endif
#endif
}

__global__ __launch_bounds__(224) void ratm_gaussian_read_kernel(
    const float* __restrict__ x,      // (B,C,H,W) f32
    const float* __restrict__ att,    // (B,G,6)   f32
    float* __restrict__ out)          // (B,G,C,GH,GW) f32
{
    __shared__ float xsA[32 * XS_STRIDE];  // X chunk buffer 0; reused as T (32 x 224)
    __shared__ float xsB[32 * XS_STRIDE];  // X chunk buffer 1 (double buffer)
    __shared__ float fyS[kH * kGH];        // Fy (224 x 32)
    __shared__ float fxS[kW * kGW];        // Fx (224 x 32)
    __shared__ float psum[7 * 1024];       // per-wave split-K partials (32x32 each)
    __shared__ float colsum[64];           // normalization sums (32 Fy + 32 Fx)

    const int tid  = threadIdx.x;          // 0..223
    const int lane = tid & 31;
    const int wv   = tid >> 5;             // 0..6
    const int l16  = lane & 15;
    const int koff = (lane >> 4) * 2;      // ISA: lanes 0-15 -> K0/K1, lanes 16-31 -> K2/K3
    const int half8 = (lane >> 4) * 8;     // C/D: lanes 16-31 hold rows M+8

    const int bid = blockIdx.x;
    const int g = bid % kG;                // g innermost -> adjacent blocks share X[b,c] in L2
    const int c = (bid / kG) % kC;
    const int b = bid / (kG * kC);

    // ---------------- Gaussian masks (per reference math) ----------------
    const float* sa = att + (size_t)(b * kG + g) * 6;
    const float cY = sa[1] * (float)kH;
    const float dY = sa[3] * ((float)(kH - 1) / (float)(kGH - 1));
    const float sY = (sa[5] + 1e-5f) * ((float)kH / (float)kGH);
    const float cX = sa[0] * (float)kW;
    const float dX = sa[2] * ((float)(kW - 1) / (float)(kGW - 1));
    const float sX = (sa[4] + 1e-5f) * ((float)kW / (float)kGW);

    {
        const float r = (float)tid;               // one len-index row per thread
        const float invSY = 1.0f / sY;
        const float invSX = 1.0f / sX;
        for (int i = 0; i < kGH; ++i) {
            float ry = (r - (cY + (float)(i - kGH / 2) * dY)) * invSY;
            fyS[tid * kGH + i] = expf(-0.5f * ry * ry);
            float rx = (r - (cX + (float)(i - kGW / 2) * dX)) * invSX;
            fxS[tid * kGW + i] = expf(-0.5f * rx * rx);
        }
    }
    __syncthreads();
    if (tid < 64) {
        float s = 0.f;
        if (tid < 32) {
            for (int r = 0; r < kH; ++r) s += fyS[r * kGH + tid];
        } else {
            const int i = tid - 32;
            for (int r = 0; r < kW; ++r) s += fxS[r * kGW + i];
        }
        colsum[tid] = s + 1e-8f;
    }
    __syncthreads();
    {
        for (int i = 0; i < kGH; ++i) {
            fyS[tid * kGH + i] /= colsum[i];
            fxS[tid * kGW + i] /= colsum[32 + i];
        }
    }
    // (mask visibility ordered by the prologue barrier below)

    // ---------------- Phase 1: T(32x224) = Fy^T (32x224) * X (224x224) ----------------
    // Wave wv owns N-strip [wv*32, wv*32+32). X is streamed HBM->LDS in 32-row chunks
    // via async DMA, double-buffered so the next chunk's DMA overlaps this chunk's WMMAs.
    const float* xp = x + ((size_t)(b * kC + c)) * (size_t)(kH * kW);
    const int n0 = wv * 32 + l16;   // global column, local n-tile 0
    const int n1 = n0 + 16;         // local n-tile 1

    v8f acc00 = {0,0,0,0,0,0,0,0};
    v8f acc01 = acc00, acc10 = acc00, acc11 = acc00;

    // prologue: stage chunk 0
    {
        const float* src = xp;                    // rows 0..31
        #pragma unroll
        for (int it = 0; it < 8; ++it) {
            int v   = tid + it * 224;             // 1792 float4 per chunk
            int row = v / (kW / 4);
            int col = (v % (kW / 4)) * 4;
            cp16_g2l(src + v * 4, &xsA[row * XS_STRIDE + col]);
        }
        async_wait0();
    }
    __syncthreads();

    for (int ch = 0; ch < 7; ++ch) {
        float* cur = (ch & 1) ? xsB : xsA;
        float* nxt = (ch & 1) ? xsA : xsB;

        // prefetch chunk ch+1 (async DMA runs under the WMMAs below)
        if (ch < 6) {
            const float* src = xp + (size_t)(ch + 1) * 32 * kW;
            #pragma unroll
            for (int it = 0; it < 8; ++it) {
                int v   = tid + it * 224;
                int row = v / (kW / 4);
                int col = (v % (kW / 4)) * 4;
                cp16_g2l(src + v * 4, &nxt[row * XS_STRIDE + col]);
            }
        }

        const int r0 = ch * 32;
        #pragma unroll
        for (int kk = 0; kk < 32; kk += 4) {
            const int kg = r0 + kk + koff;       // global r index of this lane's K pair
            v2f a0, a1, b0, b1;
            // A = Fy^T : A[m][k] = Fy[k][m]
            a0.x = fyS[kg * kGH + l16];
            a0.y = fyS[(kg + 1) * kGH + l16];
            a1.x = fyS[kg * kGH + 16 + l16];
            a1.y = fyS[(kg + 1) * kGH + 16 + l16];
            // B = X : B[k][n]
            const int kl = kk + koff;
            b0.x = cur[kl * XS_STRIDE + n0];
            b0.y = cur[(kl + 1) * XS_STRIDE + n0];
            b1.x = cur[kl * XS_STRIDE + n1];
            b1.y = cur[(kl + 1) * XS_STRIDE + n1];

            acc00 = wmma_f32_k4(a0, b0, acc00);
            acc01 = wmma_f32_k4(a0, b1, acc01);
            acc10 = wmma_f32_k4(a1, b0, acc10);
            acc11 = wmma_f32_k4(a1, b1, acc11);
        }

        if (ch < 6) async_wait0();   // my DMA for chunk ch+1 has landed
        __syncthreads();             // everyone's DMA landed; readers of 'nxt' are done
    }

    // Spill T tiles into xsA (free after the final barrier) for the w-contraction.
    #pragma unroll
    for (int v = 0; v < 8; ++v) {
        const int i0 = v + half8;
        xsA[i0 * XS_STRIDE + n0]        = acc00[v];
        xsA[i0 * XS_STRIDE + n1]        = acc01[v];
        xsA[(i0 + 16) * XS_STRIDE + n0] = acc10[v];
        xsA[(i0 + 16) * XS_STRIDE + n1] = acc11[v];
    }
    __syncthreads();

    // ---------------- Phase 2: out(32x32) = T(32x224) * Fx(224x32), split-K over waves ----------------
    v8f p00 = {0,0,0,0,0,0,0,0};
    v8f p01 = p00, p10 = p00, p11 = p00;
    const int kbase = wv * 32;
    #pragma unroll
    for (int kk = 0; kk < 32; kk += 4) {
        const int kg = kbase + kk + koff;
        v2f a0, a1, b0, b1;
        // A = T : A[m][k]  (stride 228 -> conflict-free across lane halves)
        a0.x = xsA[l16 * XS_STRIDE + kg];
        a0.y = xsA[l16 * XS_STRIDE + kg + 1];
        a1.x = xsA[(16 + l16) * XS_STRIDE + kg];
        a1.y = xsA[(16 + l16) * XS_STRIDE + kg + 1];
        // B = Fx : B[k][n]
        b0.x = fxS[kg * kGW + l16];
        b0.y = fxS[(kg + 1) * kGW + l16];
        b1.x = fxS[kg * kGW + 16 + l16];
        b1.y = fxS[(kg + 1) * kGW + 16 + l16];

        p00 = wmma_f32_k4(a0, b0, p00);
        p01 = wmma_f32_k4(a0, b1, p01);
        p10 = wmma_f32_k4(a1, b0, p10);
        p11 = wmma_f32_k4(a1, b1, p11);
    }
    {
        float* pb = &psum[wv * 1024];
        #pragma unroll
        for (int v = 0; v < 8; ++v) {
            const int i0 = v + half8;
            pb[i0 * 32 + l16]             = p00[v];
            pb[i0 * 32 + 16 + l16]        = p01[v];
            pb[(i0 + 16) * 32 + l16]      = p10[v];
            pb[(i0 + 16) * 32 + 16 + l16] = p11[v];
        }
    }
    __syncthreads();

    // Reduce the 7 split-K partials and store (B,G,C,32,32)
    float* ob = out + ((size_t)((b * kG + g) * kC + c)) * 1024;
    for (int e = tid; e < 1024; e += 224) {
        float s = 0.f;
        #pragma unroll
        for (int w2 = 0; w2 < 7; ++w2) s += psum[w2 * 1024 + e];
        ob[e] = s;
    }
}

extern "C" void kernel_launch(void* const* d_in, const int* in_sizes, int n_in,
                              void* d_out, int out_size, void* d_ws, size_t ws_size,
                              hipStream_t stream) {
    (void)in_sizes; (void)n_in; (void)out_size; (void)d_ws; (void)ws_size;
    const float* x   = (const float*)d_in[0];   // (16,64,224,224) f32
    const float* att = (const float*)d_in[1];   // (16,4,6) f32
    float* out = (float*)d_out;                 // (16,4,64,32,32) f32

    dim3 grid(kB * kG * kC);   // 4096 blocks, g innermost for L2 reuse of x[b,c]
    dim3 block(224);           // 7 wave32 waves
    ratm_gaussian_read_kernel<<<grid, block, 0, stream>>>(x, att, out);
}